// Decoder_24541443129406
// MI455X (gfx1250) — compile-verified
//
#include <hip/hip_runtime.h>

typedef __attribute__((ext_vector_type(2))) float v2f;
typedef __attribute__((ext_vector_type(8))) float v8f;

#define M_GAUSS 512
#define WPAD    12   // 9 features padded to 12 (3 x K=4 WMMA steps)

__global__ __launch_bounds__(256) void decoder_gmm_wmma(
    const float* __restrict__ origins,     // N x 2
    const float* __restrict__ directions,  // N x 2
    const float* __restrict__ latents,     // M x 6
    float* __restrict__ out)               // N x 1
{
    __shared__ float W[M_GAUSS * WPAD];    // 24 KB of the 320 KB WGP LDS

    const int tid = threadIdx.x;

    // ---- Phase A: build per-gaussian coefficient rows in LDS ----
    // exponent(n,m) = X[n,:] . W[m,:]
    // W = log2(e) * [a0..a3, b0..b3, c, 0,0,0];  a_d = 1/sigma_d,
    // b_d = mu_d/sigma_d,  c = -0.5 * sum_d mu_d^2 / sigma_d.
    // The log2(e) factor is folded into W so the inner loop can use the raw
    // v_exp_f32 (exp2) with no per-element multiply.
    const float LOG2E = 1.4426950408889634f;
    for (int m = tid; m < M_GAUSS; m += 256) {
        const float* l = latents + m * 6;
        float mu0 = l[0], mu1 = l[1], mu2 = l[2], mu3 = l[3];
        float s0 = fmaxf(l[4], 0.0f) + 0.01f;   // relu + eps
        float s1 = fmaxf(l[5], 0.0f) + 0.01f;
        float a0 = 1.0f / s0;
        float a1 = 1.0f / s1;
        const float a23 = 1000.0f;              // 1 / 0.001 (sigma_dir)
        float b0 = mu0 * a0, b1 = mu1 * a1;
        float b2 = mu2 * a23, b3 = mu3 * a23;
        float c  = -0.5f * (mu0 * b0 + mu1 * b1 + mu2 * b2 + mu3 * b3);
        float* w = &W[m * WPAD];
        w[0] = LOG2E * a0;  w[1] = LOG2E * a1;
        w[2] = LOG2E * a23; w[3] = LOG2E * a23;
        w[4] = LOG2E * b0;  w[5] = LOG2E * b1;
        w[6] = LOG2E * b2;  w[7] = LOG2E * b3;
        w[8] = LOG2E * c;   w[9] = 0.0f; w[10] = 0.0f; w[11] = 0.0f;
    }
    __syncthreads();

    // ---- Phase B: 16 points per wave x all 512 gaussians via f32 WMMA ----
    const int lane  = tid & 31;
    const int laneL = lane & 15;        // row within A / column within B
    const int h     = lane >> 4;        // half-wave selects K pair (2h, 2h+1)
    const int wave  = tid >> 5;
    const int base  = (blockIdx.x * 8 + wave) * 16;   // first point of tile
    const int p     = base + laneL;     // both halves load the same point row

    float2 o = ((const float2*)origins)[p];
    float2 d = ((const float2*)directions)[p];

    // Point feature vector X = [-.5p0^2, -.5p1^2, -.5p2^2, -.5p3^2,
    //                           p0, p1, p2, p3, 1, 0, 0, 0]
    float q0 = -0.5f * o.x * o.x;
    float q1 = -0.5f * o.y * o.y;
    float q2 = -0.5f * d.x * d.x;
    float q3 = -0.5f * d.y * d.y;

    // A-matrix (16x4 f32) chunk pairs: lane holds (K=2h, K=2h+1), branchless.
    v2f a_c0, a_c1, a_c2;
    a_c0.x = h ? q2  : q0;   a_c0.y = h ? q3  : q1;   // k 0..3
    a_c1.x = h ? d.x : o.x;  a_c1.y = h ? d.y : o.y;  // k 4..7
    a_c2.x = h ? 0.0f : 1.0f; a_c2.y = 0.0f;          // k 8..11 (9..11 = 0)

    v8f acc = {0.f, 0.f, 0.f, 0.f, 0.f, 0.f, 0.f, 0.f};

    for (int t = 0; t < 32; ++t) {
        // B-matrix (4x16 f32): lane laneL supplies gaussian g, K pair (2h,2h+1)
        int g = t * 16 + laneL;
        const float* w = &W[g * WPAD + 2 * h];
        v2f b_c0 = *(const v2f*)(w + 0);   // k base 0
        v2f b_c1 = *(const v2f*)(w + 4);   // k base 4
        v2f b_c2 = *(const v2f*)(w + 8);   // k base 8

        v8f c = {0.f, 0.f, 0.f, 0.f, 0.f, 0.f, 0.f, 0.f};
        c = __builtin_amdgcn_wmma_f32_16x16x4_f32(false, a_c0, false, b_c0,
                                                  (short)0, c, false, false);
        c = __builtin_amdgcn_wmma_f32_16x16x4_f32(false, a_c1, false, b_c1,
                                                  (short)0, c, false, false);
        c = __builtin_amdgcn_wmma_f32_16x16x4_f32(false, a_c2, false, b_c2,
                                                  (short)0, c, false, false);

        // c is already in log2 domain -> raw v_exp_f32 per element
#pragma unroll
        for (int r = 0; r < 8; ++r)
            acc[r] += __builtin_amdgcn_exp2f(c[r]);
    }

    // ---- Reduce over the 16 gaussian columns (within each half-wave) ----
#pragma unroll
    for (int r = 0; r < 8; ++r) {
        float v = acc[r];
        v += __shfl_xor(v, 1, 32);
        v += __shfl_xor(v, 2, 32);
        v += __shfl_xor(v, 4, 32);
        v += __shfl_xor(v, 8, 32);
        acc[r] = v;   // all lanes in the half now hold row sums
    }

    // Lane 0 holds points base+0..7, lane 16 holds base+8..15.
    if (laneL == 0) {
        float r0 = fminf(fmaxf(acc[0], 0.0f), 1.0f);
        float r1 = fminf(fmaxf(acc[1], 0.0f), 1.0f);
        float r2 = fminf(fmaxf(acc[2], 0.0f), 1.0f);
        float r3 = fminf(fmaxf(acc[3], 0.0f), 1.0f);
        float r4 = fminf(fmaxf(acc[4], 0.0f), 1.0f);
        float r5 = fminf(fmaxf(acc[5], 0.0f), 1.0f);
        float r6 = fminf(fmaxf(acc[6], 0.0f), 1.0f);
        float r7 = fminf(fmaxf(acc[7], 0.0f), 1.0f);
        float4* op = (float4*)&out[base + 8 * h];
        op[0] = make_float4(r0, r1, r2, r3);
        op[1] = make_float4(r4, r5, r6, r7);
    }
}

extern "C" void kernel_launch(void* const* d_in, const int* in_sizes, int n_in,
                              void* d_out, int out_size, void* d_ws, size_t ws_size,
                              hipStream_t stream) {
    (void)n_in; (void)d_ws; (void)ws_size; (void)out_size;
    const float* origins    = (const float*)d_in[0];
    const float* directions = (const float*)d_in[1];
    const float* latents    = (const float*)d_in[2];
    float* out = (float*)d_out;

    int n = in_sizes[0] / 2;            // 65536 points
    int waves  = n / 16;                // 16 points per wave
    int blocks = waves / 8;             // 8 waves per 256-thread block
    decoder_gmm_wmma<<<blocks, 256, 0, stream>>>(origins, directions, latents, out);
}